// GraphConvolution_55070070670123
// MI455X (gfx1250) — compile-verified
//
#include <hip/hip_runtime.h>
#include <hip/hip_bf16.h>

typedef float v2f __attribute__((ext_vector_type(2)));
typedef float v8f __attribute__((ext_vector_type(8)));

#define N_NODES 100000
#define N_EDGES 1600000
#define IN_DIM  128
#define OUT_DIM 64
#define CHUNK   1024

// ---------------------------------------------------------------------------
// GEMM: H[N,64] = X[N,128] @ W[128,64] using V_WMMA_F32_16X16X4_F32.
// One wave computes a 16x64 strip (4 accumulators of 16x16), K in steps of 4.
// A frag (16x4): lanes 0-15 = M rows, VGPR0/1 = K0/K1; lanes 16-31 = K2/K3.
// B frag (4x16): lanes 0-15 = N cols, VGPR0/1 = K0/K1; lanes 16-31 = K2/K3.
// C/D (16x16):   lanes 0-15 = N cols with M=r, lanes 16-31 with M=r+8.
// ---------------------------------------------------------------------------
__global__ void gemm_wmma_f32(const float* __restrict__ X,
                              const float* __restrict__ W,
                              float* __restrict__ H, int nTiles) {
    int wave = (blockIdx.x * blockDim.x + threadIdx.x) >> 5;
    int lane = threadIdx.x & 31;
    if (wave >= nTiles) return;                 // wave-uniform: EXEC stays all-1s

    int row0  = wave * 16;
    int m     = lane & 15;
    int khalf = lane >> 4;                      // 0 or 1

    v8f acc0 = {}, acc1 = {}, acc2 = {}, acc3 = {};
    const float* xrow = X + (long)(row0 + m) * IN_DIM;

    for (int k = 0; k < IN_DIM; k += 4) {
        int ka = k + khalf * 2;
        v2f a;
        a.x = xrow[ka];
        a.y = xrow[ka + 1];
        const float* w0 = W + (long)ka * OUT_DIM;
        const float* w1 = w0 + OUT_DIM;
        v2f b0, b1, b2, b3;
        b0.x = w0[m];      b0.y = w1[m];
        b1.x = w0[m + 16]; b1.y = w1[m + 16];
        b2.x = w0[m + 32]; b2.y = w1[m + 32];
        b3.x = w0[m + 48]; b3.y = w1[m + 48];
        acc0 = __builtin_amdgcn_wmma_f32_16x16x4_f32(false, a, false, b0, (short)0, acc0, false, false);
        acc1 = __builtin_amdgcn_wmma_f32_16x16x4_f32(false, a, false, b1, (short)0, acc1, false, false);
        acc2 = __builtin_amdgcn_wmma_f32_16x16x4_f32(false, a, false, b2, (short)0, acc2, false, false);
        acc3 = __builtin_amdgcn_wmma_f32_16x16x4_f32(false, a, false, b3, (short)0, acc3, false, false);
    }

    float* hrow = H + (long)row0 * OUT_DIM;
    int rbase = khalf * 8;
#pragma unroll
    for (int r = 0; r < 8; ++r) {
        int row = rbase + r;
        hrow[row * OUT_DIM + m]      = acc0[r];
        hrow[row * OUT_DIM + m + 16] = acc1[r];
        hrow[row * OUT_DIM + m + 32] = acc2[r];
        hrow[row * OUT_DIM + m + 48] = acc3[r];
    }
}

// ---------------------------------------------------------------------------
// CSR build (once per launch): counts -> exclusive scan -> stable-ish scatter
// ---------------------------------------------------------------------------
__global__ void zero_i32(int* __restrict__ p, int n) {
    int i = blockIdx.x * blockDim.x + threadIdx.x;
    if (i < n) p[i] = 0;
}

__global__ void count_rows(const int* __restrict__ erow, int* __restrict__ counts, int E) {
    int i = blockIdx.x * blockDim.x + threadIdx.x;
    if (i < E) atomicAdd(&counts[erow[i]], 1);
}

__global__ void chunk_sums(const int* __restrict__ counts, int* __restrict__ sums, int n) {
    __shared__ int sdata[256];
    int base = blockIdx.x * CHUNK;
    int t = threadIdx.x;
    int s = 0;
    for (int i = t; i < CHUNK; i += 256) {
        int idx = base + i;
        if (idx < n) s += counts[idx];
    }
    sdata[t] = s;
    __syncthreads();
    for (int off = 128; off > 0; off >>= 1) {
        if (t < off) sdata[t] += sdata[t + off];
        __syncthreads();
    }
    if (t == 0) sums[blockIdx.x] = sdata[0];
}

__global__ void scan_sums(const int* __restrict__ sums, int* __restrict__ offsets, int nb) {
    int run = 0;
    for (int i = 0; i < nb; ++i) { offsets[i] = run; run += sums[i]; }
}

__global__ void scan_write(const int* __restrict__ counts, const int* __restrict__ offsets,
                           int* __restrict__ rowStart, int* __restrict__ cursor, int n) {
    if (threadIdx.x != 0) return;
    int base = blockIdx.x * CHUNK;
    int run  = offsets[blockIdx.x];
    int end  = base + CHUNK; if (end > n) end = n;
    for (int i = base; i < end; ++i) {
        rowStart[i] = run;
        cursor[i]   = run;
        run += counts[i];
    }
    if (end == n) rowStart[n] = run;
}

__global__ void scatter_edges(const int* __restrict__ erow, int* __restrict__ cursor,
                              int* __restrict__ perm, int E) {
    int i = blockIdx.x * blockDim.x + threadIdx.x;
    if (i < E) {
        int p = atomicAdd(&cursor[erow[i]], 1);
        perm[p] = i;
    }
}

// ---------------------------------------------------------------------------
// One propagation hop: hout[r] = 0.5*(sum_e val_e * hin[col_e] + hin[r]).
// One wave per row; each lane owns a float2 (2 of 64 columns). Lanes
// cooperatively load up to 32 edges then broadcast (col,val) via __shfl.
// Final hop fuses bias + relu and writes d_out.
// ---------------------------------------------------------------------------
__global__ void spmm_hop(const float* __restrict__ hin, float* __restrict__ hout,
                         const int* __restrict__ rowStart, const int* __restrict__ perm,
                         const float* __restrict__ eval, const int* __restrict__ ecol,
                         const float* __restrict__ bias, int nRows, int finalHop) {
    int row  = (blockIdx.x * blockDim.x + threadIdx.x) >> 5;
    int lane = threadIdx.x & 31;
    if (row >= nRows) return;

    int s = rowStart[row];
    int e = rowStart[row + 1];
    float2 acc = make_float2(0.f, 0.f);

    for (int j = s; j < e; j += 32) {
        int n = e - j; if (n > 32) n = 32;
        int   myC = 0; float myV = 0.f;
        if (lane < n) {
            int id = perm[j + lane];
            myC = ecol[id];
            myV = eval[id];
        }
        for (int i = 0; i < n; ++i) {
            int   c = __shfl(myC, i, 32);
            float v = __shfl(myV, i, 32);
            float2 hv = ((const float2*)(hin + (long)c * OUT_DIM))[lane];
            acc.x += v * hv.x;
            acc.y += v * hv.y;
        }
    }

    float2 hs = ((const float2*)(hin + (long)row * OUT_DIM))[lane];
    float2 o;
    o.x = 0.5f * (acc.x + hs.x);
    o.y = 0.5f * (acc.y + hs.y);
    if (finalHop) {
        float2 bv = ((const float2*)bias)[lane];
        o.x = fmaxf(o.x + bv.x, 0.f);
        o.y = fmaxf(o.y + bv.y, 0.f);
    }
    ((float2*)(hout + (long)row * OUT_DIM))[lane] = o;
}

// ---------------------------------------------------------------------------
extern "C" void kernel_launch(void* const* d_in, const int* in_sizes, int n_in,
                              void* d_out, int out_size, void* d_ws, size_t ws_size,
                              hipStream_t stream) {
    const float* x     = (const float*)d_in[0];
    const float* w     = (const float*)d_in[1];
    const float* bias  = (const float*)d_in[2];
    const float* eval  = (const float*)d_in[3];
    const int*   erow  = (const int*)d_in[4];
    const int*   ecol  = (const int*)d_in[5];
    float* out = (float*)d_out;

    const int N = N_NODES, E = N_EDGES;
    const int NB = (N + CHUNK - 1) / CHUNK;          // scan chunks

    // workspace carve-up (all 256B-aligned)
    char* p = (char*)d_ws;
    auto take = [&](size_t bytes) { char* q = p; p += (bytes + 255) & ~size_t(255); return q; };
    float* hA       = (float*)take((size_t)N * OUT_DIM * sizeof(float));
    float* hB       = (float*)take((size_t)N * OUT_DIM * sizeof(float));
    int*   counts   = (int*)  take((size_t)N * sizeof(int));
    int*   rowStart = (int*)  take((size_t)(N + 1) * sizeof(int));
    int*   cursor   = (int*)  take((size_t)N * sizeof(int));
    int*   perm     = (int*)  take((size_t)E * sizeof(int));
    int*   sums     = (int*)  take((size_t)NB * sizeof(int));
    int*   offsets  = (int*)  take((size_t)NB * sizeof(int));
    (void)ws_size;

    // 1) GEMM via fp32 WMMA: 6250 tiles of 16 rows, 8 waves (tiles) per block
    {
        int nTiles = N / 16;                          // 100000/16 = 6250
        int blocks = (nTiles + 7) / 8;
        gemm_wmma_f32<<<blocks, 256, 0, stream>>>(x, w, hA, nTiles);
    }

    // 2) CSR build (recomputed each launch; deterministic work)
    zero_i32<<<(N + 255) / 256, 256, 0, stream>>>(counts, N);
    count_rows<<<(E + 255) / 256, 256, 0, stream>>>(erow, counts, E);
    chunk_sums<<<NB, 256, 0, stream>>>(counts, sums, N);
    scan_sums<<<1, 1, 0, stream>>>(sums, offsets, NB);
    scan_write<<<NB, 32, 0, stream>>>(counts, offsets, rowStart, cursor, N);
    scatter_edges<<<(E + 255) / 256, 256, 0, stream>>>(erow, cursor, perm, E);

    // 3) 4 hops, atomic-free gather; ping-pong hA <-> hB; last hop -> d_out
    int blocks = (N * 32 + 255) / 256;                // one wave per row
    spmm_hop<<<blocks, 256, 0, stream>>>(hA, hB, rowStart, perm, eval, ecol, bias, N, 0);
    spmm_hop<<<blocks, 256, 0, stream>>>(hB, hA, rowStart, perm, eval, ecol, bias, N, 0);
    spmm_hop<<<blocks, 256, 0, stream>>>(hA, hB, rowStart, perm, eval, ecol, bias, N, 0);
    spmm_hop<<<blocks, 256, 0, stream>>>(hB, out, rowStart, perm, eval, ecol, bias, N, 1);
}